// C_TABLE_ALL_29987461660960
// MI455X (gfx1250) — compile-verified
//
#include <hip/hip_runtime.h>

#define NDIM   2048
#define KSTEPS 32
#define BLOCK  256
#define NWAVES (BLOCK / 32)
#define PER_T  (NDIM / BLOCK)   // 8 elements per thread
#define BIGF   1e30f

// address-space helper types for the gfx1250 async global->LDS builtins
typedef __attribute__((address_space(1))) void GV;  // global
typedef __attribute__((address_space(3))) void LV;  // LDS
typedef int v4i __attribute__((ext_vector_type(4)));
typedef __attribute__((address_space(1))) v4i* gv4i_p;
typedef __attribute__((address_space(3))) v4i* lv4i_p;

// Stage 16 bytes (4 floats) from global into LDS via the CDNA5 async DMA path
// (tracked by ASYNCcnt, no VGPR data movement).
__device__ __forceinline__ void async_copy16(const float* g, float* l) {
#if __has_builtin(__builtin_amdgcn_global_load_async_to_lds_b128)
    gv4i_p gp = (gv4i_p)((GV*)g);
    lv4i_p lp = (lv4i_p)((LV*)l);
    __builtin_amdgcn_global_load_async_to_lds_b128(gp, lp, /*offset=*/0, /*cpol=*/0);
#else
    unsigned lds_addr = (unsigned)(unsigned long long)(LV*)l;
    asm volatile("global_load_async_to_lds_b128 %0, %1, off"
                 :: "v"(lds_addr), "v"(g) : "memory");
#endif
}

__device__ __forceinline__ void wait_async_zero() {
    asm volatile("s_wait_asynccnt 0" ::: "memory");
}

// wave32 butterfly reductions
__device__ __forceinline__ float wave_min(float v) {
#pragma unroll
    for (int off = 16; off > 0; off >>= 1) v = fminf(v, __shfl_xor(v, off, 32));
    return v;
}
__device__ __forceinline__ float wave_sum(float v) {
#pragma unroll
    for (int off = 16; off > 0; off >>= 1) v += __shfl_xor(v, off, 32);
    return v;
}

// One DP step kk (1..31). One block per row n.
__global__ __launch_bounds__(BLOCK) void ctable_step(
    const float* __restrict__ D,
    const float* __restrict__ cprev,
    float* __restrict__ cnext,
    float* __restrict__ Cout,
    float* __restrict__ Call,
    int kk)
{
    __shared__ float ldsD[NDIM];
    __shared__ float ldsC[NDIM + 4];   // [0..NDIM-1] = cprev, [NDIM] = 0 pad
    __shared__ float redmin[NWAVES];
    __shared__ float redsum[NWAVES];

    const int n = blockIdx.x;
    const int t = threadIdx.x;
    const int wid = t >> 5;
    const int lane = t & 31;
    const int limit = NDIM - kk;       // valid i: n <= i < limit

    const float* drow = D + (size_t)n * NDIM;

    // Async-stage D row and cprev into LDS (4 x b128 per thread, fully coalesced).
    {
        const int e0 = t * 4;
        async_copy16(drow  + e0,             &ldsD[e0]);
        async_copy16(drow  + e0 + BLOCK * 4, &ldsD[e0 + BLOCK * 4]);
        async_copy16(cprev + e0,             &ldsC[e0]);
        async_copy16(cprev + e0 + BLOCK * 4, &ldsC[e0 + BLOCK * 4]);
    }
    if (t == 0) ldsC[NDIM] = 0.0f;     // C_prev[N] := 0 (shift pad)
    wait_async_zero();
    __syncthreads();

    const int base = t * PER_T;        // contiguous 8-element chunk per thread

    // Pass 1: M = D + shift(C_prev); masked min.
    float Mv[PER_T];
    float m = BIGF;
#pragma unroll
    for (int j = 0; j < PER_T; ++j) {
        const int i = base + j;
        const float Mval = ldsD[i] + ldsC[i + 1];
        Mv[j] = Mval;
        const bool valid = (i >= n) & (i < limit);
        m = valid ? fminf(m, Mval) : m;
    }
    m = wave_min(m);
    if (lane == 0) redmin[wid] = m;
    __syncthreads();
    float mall = redmin[0];
#pragma unroll
    for (int w = 1; w < NWAVES; ++w) mall = fminf(mall, redmin[w]);

    // Pass 2: exp(m - M) over valid window; masked sum (invalid terms are exactly 0,
    // matching the reference's exp(-BIG - max) underflow).
    float ev[PER_T];
    float ssum = 0.0f;
#pragma unroll
    for (int j = 0; j < PER_T; ++j) {
        const int i = base + j;
        const bool valid = (i >= n) & (i < limit);
        const float e = valid ? __expf(mall - Mv[j]) : 0.0f;
        ev[j] = e;
        ssum += e;
    }
    ssum = wave_sum(ssum);
    if (lane == 0) redsum[wid] = ssum;
    __syncthreads();
    float sall = redsum[0];
#pragma unroll
    for (int w = 1; w < NWAVES; ++w) sall += redsum[w];

    // Pass 3: normalized softmax (or -1 outside window), 16B vector stores.
    const float inv = (sall > 0.0f) ? (1.0f / sall) : 0.0f;
    float* orow = Call + ((size_t)n * KSTEPS + kk) * NDIM;
#pragma unroll
    for (int g = 0; g < PER_T / 4; ++g) {
        const int i0 = base + g * 4;
        float4 o;
        o.x = ((i0 + 0 >= n) & (i0 + 0 < limit)) ? ev[g * 4 + 0] * inv : -1.0f;
        o.y = ((i0 + 1 >= n) & (i0 + 1 < limit)) ? ev[g * 4 + 1] * inv : -1.0f;
        o.z = ((i0 + 2 >= n) & (i0 + 2 < limit)) ? ev[g * 4 + 2] * inv : -1.0f;
        o.w = ((i0 + 3 >= n) & (i0 + 3 < limit)) ? ev[g * 4 + 3] * inv : -1.0f;
        *(float4*)(orow + i0) = o;
    }

    if (t == 0) {
        const float ck = (n < limit) ? mall : 0.0f;   // row_valid ? min : 0
        Cout[n * KSTEPS + kk] = ck;
        cnext[n] = ck;
    }
}

// kk = 0 initialization: C[:,0] = D[:,N-1]; C_all[:,0,:] = -1 except col N-1 = 1.
__global__ __launch_bounds__(BLOCK) void ctable_init(
    const float* __restrict__ D, float* __restrict__ Cout,
    float* __restrict__ Call, float* __restrict__ c0buf)
{
    const int n = blockIdx.x;
    const int t = threadIdx.x;
    float4* orow = (float4*)(Call + (size_t)n * KSTEPS * NDIM);  // row (n, kk=0)
    for (int j = t; j < NDIM / 4; j += BLOCK) {
        float4 v = make_float4(-1.0f, -1.0f, -1.0f, -1.0f);
        if (j == NDIM / 4 - 1) v.w = 1.0f;   // element i = N-1
        orow[j] = v;
    }
    if (t == 0) {
        const float c0 = D[(size_t)n * NDIM + (NDIM - 1)];
        Cout[n * KSTEPS] = c0;
        c0buf[n] = c0;
    }
}

extern "C" void kernel_launch(void* const* d_in, const int* in_sizes, int n_in,
                              void* d_out, int out_size, void* d_ws, size_t ws_size,
                              hipStream_t stream) {
    (void)in_sizes; (void)n_in; (void)out_size; (void)ws_size;
    const float* D    = (const float*)d_in[0];
    float*       C    = (float*)d_out;                              // (N, K)
    float*       Call = (float*)d_out + (size_t)NDIM * KSTEPS;      // (N, K, N)
    float*       bufA = (float*)d_ws;                               // ping
    float*       bufB = bufA + NDIM;                                // pong

    ctable_init<<<NDIM, BLOCK, 0, stream>>>(D, C, Call, bufA);

    float* prev = bufA;
    float* next = bufB;
    for (int kk = 1; kk < KSTEPS; ++kk) {
        ctable_step<<<NDIM, BLOCK, 0, stream>>>(D, prev, next, C, Call, kk);
        float* tmp = prev; prev = next; next = tmp;
    }
}